// DetectionTargetLayer_22849226015387
// MI455X (gfx1250) — compile-verified
//
#include <hip/hip_runtime.h>
#include <stdint.h>

#define R_PROP   6000
#define T_TRUE   100
#define T_PAD    112
#define N_IMG    32
#define ROI_N    512
#define POS_MAX  128
#define THRESH   0.5f
#define MASK_W   ((R_PROP + 31) / 32)   // 188

typedef float v2f __attribute__((ext_vector_type(2)));
typedef float v8f __attribute__((ext_vector_type(8)));

// ---- CDNA5 async global -> LDS staging -------------------------------------
__device__ __forceinline__ uint32_t lds_off_u32(const void* p) {
  // addrspace(3) -> flat keeps the LDS byte offset in the low 32 bits
  return (uint32_t)(uintptr_t)p;
}
__device__ __forceinline__ void async_copy_b128(uint32_t ldsOff, const void* g) {
  asm volatile("global_load_async_to_lds_b128 %0, %1, off"
               :: "v"(ldsOff), "v"(g) : "memory");
}
__device__ __forceinline__ void wait_async0() {
  asm volatile("s_wait_asynccnt 0" ::: "memory");
}

// ---- order-preserving (iou, first-index) packing for 64-bit max ------------
__device__ __forceinline__ unsigned long long packIou(float iou, unsigned r) {
  unsigned u = __float_as_uint(iou);
  u = (u & 0x80000000u) ? ~u : (u | 0x80000000u);
  return ((unsigned long long)u << 32) |
         (unsigned long long)(0xFFFFFFFFu - r);   // ties -> smaller r wins
}
__device__ __forceinline__ unsigned long long shflxor_u64(unsigned long long v, int m) {
  unsigned lo = (unsigned)v, hi = (unsigned)(v >> 32);
  lo = __shfl_xor(lo, m);
  hi = __shfl_xor(hi, m);
  return ((unsigned long long)hi << 32) | lo;
}

__global__ void init_best(unsigned long long* b, int n) {
  int i = blockIdx.x * blockDim.x + threadIdx.x;
  if (i < n) b[i] = 0ull;
}

// ---- Pass 1: IoU tiles. WMMA computes area_p[m]+area_t[n]; VALU does -------
// ---- the min/max intersection; shuffle+LDS u64-max reductions. -------------
__global__ __launch_bounds__(256) void iou_pass1(
    const float* __restrict__ prop, const float* __restrict__ tb,
    float* __restrict__ iouMax, unsigned long long* __restrict__ gBest)
{
  __shared__ float sT[T_PAD * 4];
  __shared__ float sP[128 * 4];
  __shared__ float sAreaT[T_PAD];
  __shared__ int   sValidT[T_PAD];
  __shared__ unsigned long long sBest[T_PAD];

  const int img = blockIdx.y;
  const int tid = threadIdx.x;
  const int blockBase = blockIdx.x * 128;

  for (int i = tid; i < T_PAD; i += 256) sBest[i] = 0ull;
  if (tid < T_PAD - T_TRUE) {                 // zero-pad truths 100..111
    int t = T_TRUE + tid;
    sT[t*4+0] = sT[t*4+1] = sT[t*4+2] = sT[t*4+3] = 0.f;
  }
  if (tid < T_TRUE)                           // async stage 100 truth boxes
    async_copy_b128(lds_off_u32(&sT[tid*4]),
                    tb + (size_t)(img * T_TRUE + tid) * 4);
  if (tid < 128) {                            // async stage 128 proposal rows
    int r = blockBase + tid;
    if (r < R_PROP)
      async_copy_b128(lds_off_u32(&sP[tid*4]),
                      prop + (size_t)img * R_PROP * 4 + (size_t)r * 4);
    else
      sP[tid*4+0] = sP[tid*4+1] = sP[tid*4+2] = sP[tid*4+3] = 0.f;
  }
  wait_async0();
  __syncthreads();

  if (tid < T_PAD) {
    float y1 = sT[tid*4], x1 = sT[tid*4+1], y2 = sT[tid*4+2], x2 = sT[tid*4+3];
    sAreaT[tid]  = (y2 - y1) * (x2 - x1);
    sValidT[tid] = (y1 != 0.f) | (x1 != 0.f) | (y2 != 0.f) | (x2 != 0.f);
  }
  __syncthreads();

  const int wave = tid >> 5, lane = tid & 31;
  const int hlf = lane >> 4, ln = lane & 15;
  const int waveRow = wave * 16;              // 16 proposals per wave
  const float hsel = (hlf == 0) ? 1.f : 0.f;  // lane-select, no EXEC churn

  // A fragment (16x4 f32): col0 = area_p, col1 = 1, cols 2..3 = 0
  v2f a;
  {
    const float* pb = &sP[(waveRow + ln) * 4];
    float areaP = (pb[2] - pb[0]) * (pb[3] - pb[1]);
    a.x = areaP * hsel;                       // K=0 (lanes 0-15) / K=2 = 0
    a.y = hsel;                               // K=1 (lanes 0-15) / K=3 = 0
  }

  // hoist this lane's 8 proposal boxes (rows m = 8*hlf + k) into registers
  float P0[8], P1[8], P2[8], P3[8];
  bool  VP[8];
  #pragma unroll
  for (int k = 0; k < 8; ++k) {
    const float* pb = &sP[(waveRow + 8 * hlf + k) * 4];
    P0[k] = pb[0]; P1[k] = pb[1]; P2[k] = pb[2]; P3[k] = pb[3];
    VP[k] = (P0[k] != 0.f) | (P1[k] != 0.f) | (P2[k] != 0.f) | (P3[k] != 0.f);
  }

  float rm[8];
  #pragma unroll
  for (int k = 0; k < 8; ++k) rm[k] = -2.f;

  #pragma unroll
  for (int tile = 0; tile < 7; ++tile) {
    const int nIdx = tile * 16 + ln;
    // B fragment (4x16 f32): row0 = 1, row1 = area_t, rows 2..3 = 0
    v2f b;
    b.x = hsel;
    b.y = sAreaT[nIdx] * hsel;
    v8f c = {};
    // d[m,n] = area_p[m] + area_t[n]   (v_wmma_f32_16x16x4_f32)
    v8f d = __builtin_amdgcn_wmma_f32_16x16x4_f32(
        false, a, false, b, (short)0, c, false, false);

    const float ty1 = sT[nIdx*4+0], tx1 = sT[nIdx*4+1];
    const float ty2 = sT[nIdx*4+2], tx2 = sT[nIdx*4+3];
    const bool  vt  = sValidT[nIdx] != 0;

    unsigned long long colKey = 0ull;
    #pragma unroll
    for (int k = 0; k < 8; ++k) {
      float ih = fminf(P2[k], ty2) - fmaxf(P0[k], ty1);
      float iw = fminf(P3[k], tx2) - fmaxf(P1[k], tx1);
      float inter = fmaxf(ih, 0.f) * fmaxf(iw, 0.f);
      float uni = d[k] - inter;               // union from WMMA accumulator
      // fast reciprocal: IoU feeds comparisons only (v_rcp_f32, 1 trans op)
      float iou = (VP[k] && vt && uni > 0.f)
                    ? inter * __builtin_amdgcn_rcpf(uni) : -1.f;
      rm[k] = fmaxf(rm[k], iou);
      unsigned r = (unsigned)(blockBase + waveRow + 8 * hlf + k);
      unsigned long long key = (iou >= 0.f) ? packIou(iou, r) : 0ull;
      colKey = (key > colKey) ? key : colKey;  // branchless
    }
    unsigned long long other = shflxor_u64(colKey, 16);  // fold the two halves
    colKey = (other > colKey) ? other : colKey;
    if (hlf == 0 && colKey) atomicMax(&sBest[nIdx], colKey);  // ds_max_u64
  }

  // row max: reduce across the 16 lanes sharing each row
  #pragma unroll
  for (int k = 0; k < 8; ++k) {
    float v = rm[k];
    v = fmaxf(v, __shfl_xor(v, 1));
    v = fmaxf(v, __shfl_xor(v, 2));
    v = fmaxf(v, __shfl_xor(v, 4));
    v = fmaxf(v, __shfl_xor(v, 8));
    rm[k] = v;
  }
  if (ln == 0) {
    #pragma unroll
    for (int k = 0; k < 8; ++k) {
      int r = blockBase + waveRow + 8 * hlf + k;
      if (r < R_PROP) iouMax[(size_t)img * R_PROP + r] = rm[k];
    }
  }
  __syncthreads();
  if (tid < T_TRUE) {
    unsigned long long k = sBest[tid];
    if (k) atomicMax(&gBest[img * T_TRUE + tid], k);
  }
}

// ---- Pass 2: per-image deterministic sampling + output ---------------------
__global__ __launch_bounds__(32) void sample_pass2(
    const float* __restrict__ prop, const int* __restrict__ tcls,
    const float* __restrict__ tb, const float* __restrict__ iouMax,
    const unsigned long long* __restrict__ gBest, float* __restrict__ out)
{
  __shared__ unsigned forcedMask[MASK_W];
  __shared__ float sT[T_TRUE * 4];
  __shared__ int   sValidT[T_TRUE];
  __shared__ int   posList[POS_MAX];
  __shared__ int   negList[ROI_N];

  const int img = blockIdx.x;
  const int lane = threadIdx.x;

  for (int i = lane; i < MASK_W; i += 32) forcedMask[i] = 0u;
  for (int t = lane; t < T_TRUE; t += 32)
    async_copy_b128(lds_off_u32(&sT[t*4]),
                    tb + (size_t)(img * T_TRUE + t) * 4);
  wait_async0();
  __syncthreads();

  for (int t = lane; t < T_TRUE; t += 32) {
    float y1 = sT[t*4], x1 = sT[t*4+1], y2 = sT[t*4+2], x2 = sT[t*4+3];
    int v = (y1 != 0.f) | (x1 != 0.f) | (y2 != 0.f) | (x2 != 0.f);
    sValidT[t] = v;
    unsigned long long k = gBest[img * T_TRUE + t];
    if (v && k) {
      unsigned r = 0xFFFFFFFFu - (unsigned)(k & 0xFFFFFFFFull);
      if (r < (unsigned)R_PROP) atomicOr(&forcedMask[r >> 5], 1u << (r & 31));
    }
  }
  __syncthreads();

  // deterministic ordered compaction of positives / negatives (single wave)
  int posCount = 0, negCount = 0;
  for (int base = 0; base < R_PROP; base += 32) {
    int r = base + lane;
    bool ok = r < R_PROP;
    float im = ok ? iouMax[(size_t)img * R_PROP + r] : -2.f;
    bool forced = ok && ((forcedMask[r >> 5] >> (r & 31)) & 1u);
    bool vp = false;
    if (ok) {
      const float* pb = prop + (size_t)img * R_PROP * 4 + (size_t)r * 4;
      vp = (pb[0] != 0.f) | (pb[1] != 0.f) | (pb[2] != 0.f) | (pb[3] != 0.f);
    }
    bool isPos = vp && ((im >= THRESH) || forced);
    bool isNeg = vp && !isPos && (im < THRESH);
    unsigned long long bp = __ballot(isPos);
    unsigned long long bn = __ballot(isNeg);
    unsigned long long lt = (1ull << lane) - 1ull;
    if (isPos) { int s = posCount + (int)__popcll(bp & lt);
                 if (s < POS_MAX) posList[s] = r; }
    if (isNeg) { int s = negCount + (int)__popcll(bn & lt);
                 if (s < ROI_N) negList[s] = r; }
    posCount += (int)__popcll(bp);
    negCount += (int)__popcll(bn);
  }
  __syncthreads();
  int p = posCount < POS_MAX ? posCount : POS_MAX;
  int nn = ROI_N - p;
  if (negCount < nn) nn = negCount;

  float* roiOut = out + (size_t)img * ROI_N * 4;
  float* clsOut = out + (size_t)N_IMG * ROI_N * 4 + (size_t)img * ROI_N;
  float* delOut = out + (size_t)N_IMG * ROI_N * 4 + (size_t)N_IMG * ROI_N
                      + (size_t)img * ROI_N * 4;

  for (int s = lane; s < ROI_N; s += 32) {
    float r0 = 0.f, r1 = 0.f, r2 = 0.f, r3 = 0.f;
    float d0 = 0.f, d1 = 0.f, d2 = 0.f, d3 = 0.f;
    float cls = -1.f;
    if (s < p + nn) {
      int r = (s < p) ? posList[s] : negList[s - p];
      const float* pb = prop + (size_t)img * R_PROP * 4 + (size_t)r * 4;
      float py1 = pb[0], px1 = pb[1], py2 = pb[2], px2 = pb[3];
      r0 = py1; r1 = px1; r2 = py2; r3 = px2;
      if (s < p) {
        float areaP = (py2 - py1) * (px2 - px1);
        float best = -1e30f; int bt = 0;
        for (int t = 0; t < T_TRUE; ++t) {
          float ty1 = sT[t*4], tx1 = sT[t*4+1], ty2 = sT[t*4+2], tx2 = sT[t*4+3];
          float inter = fmaxf(fminf(py2, ty2) - fmaxf(py1, ty1), 0.f) *
                        fmaxf(fminf(px2, tx2) - fmaxf(px1, tx1), 0.f);
          float at = (ty2 - ty1) * (tx2 - tx1);
          float uni = areaP + at - inter;
          float iou = (sValidT[t] && uni > 0.f)
                        ? inter * __builtin_amdgcn_rcpf(uni) : -1.f;
          if (iou > best) { best = iou; bt = t; }   // first-max tie-break
        }
        float ty1 = sT[bt*4], tx1 = sT[bt*4+1], ty2 = sT[bt*4+2], tx2 = sT[bt*4+3];
        float ph = py2 - py1, pw = px2 - px1;
        float th = ty2 - ty1, tw = tx2 - tx1;
        float pcy = py1 + 0.5f * ph, pcx = px1 + 0.5f * pw;
        float tcy = ty1 + 0.5f * th, tcx = tx1 + 0.5f * tw;
        d0 = (tcy - pcy) / ph; d1 = (tcx - pcx) / pw;   // precise for outputs
        d2 = logf(th / ph);    d3 = logf(tw / pw);
        cls = (float)tcls[img * T_TRUE + bt];
      } else {
        cls = 0.f;
      }
    }
    roiOut[s*4+0] = r0; roiOut[s*4+1] = r1; roiOut[s*4+2] = r2; roiOut[s*4+3] = r3;
    clsOut[s] = cls;
    delOut[s*4+0] = d0; delOut[s*4+1] = d1; delOut[s*4+2] = d2; delOut[s*4+3] = d3;
  }
}

extern "C" void kernel_launch(void* const* d_in, const int* in_sizes, int n_in,
                              void* d_out, int out_size, void* d_ws, size_t ws_size,
                              hipStream_t stream) {
  (void)in_sizes; (void)n_in; (void)out_size; (void)ws_size;
  const float* prop = (const float*)d_in[0];          // [32,6000,4] f32
  const int*   tcls = (const int*)d_in[1];            // [32,100]    i32
  const float* tb   = (const float*)d_in[2];          // [32,100,4]  f32
  float* out = (float*)d_out;                         // roi | cls | deltas

  float* iouMax = (float*)d_ws;                       // 32*6000 f32
  unsigned long long* gBest = (unsigned long long*)
      ((char*)d_ws + (size_t)N_IMG * R_PROP * sizeof(float));  // 32*100 u64

  hipLaunchKernelGGL(init_best, dim3((N_IMG * T_TRUE + 255) / 256), dim3(256),
                     0, stream, gBest, N_IMG * T_TRUE);
  dim3 g1((R_PROP + 127) / 128, N_IMG);
  hipLaunchKernelGGL(iou_pass1, g1, dim3(256), 0, stream, prop, tb, iouMax, gBest);
  hipLaunchKernelGGL(sample_pass2, dim3(N_IMG), dim3(32), 0, stream,
                     prop, tcls, tb, iouMax, gBest, out);
}